// HashGridEncoder2D_17678085390370
// MI455X (gfx1250) — compile-verified
//
#include <hip/hip_runtime.h>
#include <stdint.h>

// HashGridEncoder2D for MI455X (gfx1250, wave32).
// - 1 thread = 1 point; 256-thread tile.
// - Levels processed in groups of 4: 16 independent global_load_b64 gathers
//   issued per batch (all addresses depend only on ux/uy), then consumed ->
//   high per-wave MLP, few waitcnt stalls. Table (64 MiB) is L2-resident.
// - Results staged in LDS (point-major, 144B stride, 16B aligned), then
//   DMA'd out with CDNA5 async LDS->global b128 stores (ASYNCcnt) using a
//   non-temporal hint so the 256 MiB output stream doesn't evict the table
//   from L2. Output writes are perfectly coalesced (512B/wave).

#define HG_LEVELS 16
#define HG_TABLE  (1u << 19)
#define HG_TMASK  (HG_TABLE - 1u)
#define HG_HA     73856093u
#define HG_HB     19349663u

// Per-point LDS stride in floats: 32 features + 4 pad -> 144 bytes (16B aligned).
#define HG_STRIDE 36

__global__ __launch_bounds__(256) void hashgrid2d_encode_kernel(
    const float* __restrict__ xy, const float* __restrict__ emb,
    float* __restrict__ out, int n)
{
    // int(16 * 1.5**l) for l in 0..15 (exact in double)
    constexpr int kRes[HG_LEVELS] = {16, 24, 36, 54, 81, 121, 182, 273,
                                     410, 615, 922, 1383, 2075, 3113, 4670, 7006};

    __shared__ float lds[256 * HG_STRIDE];   // 36 KB / block

    const int tid  = (int)threadIdx.x;
    const int tile = (int)blockIdx.x;
    const int p    = tile * 256 + tid;

    float ux = 0.0f, uy = 0.0f;
    if (p < n) {
        const float2 c = *(const float2*)(xy + 2 * (size_t)p);   // coalesced b64
        ux = fminf(fmaxf((c.x + 1.0f) * 0.5f, 0.0f), 1.0f);      // v_max_num/v_min_num
        uy = fminf(fmaxf((c.y + 1.0f) * 0.5f, 0.0f), 1.0f);
    }

    const float2* __restrict__ emb2 = (const float2*)emb;

#pragma unroll
    for (int g = 0; g < HG_LEVELS / 4; ++g) {
        float2 e[4][4];
        float  wxv[4], wyv[4];

        // --- Phase A: compute 16 hash indices, issue 16 independent gathers ---
#pragma unroll
        for (int i = 0; i < 4; ++i) {
            const int   l   = g * 4 + i;
            const int   res = kRes[l];
            const float rm1 = (float)(res - 1);

            const float px = ux * rm1;
            const float py = uy * rm1;
            const int   x0 = (int)px;           // px >= 0 -> trunc == floor
            const int   y0 = (int)py;
            wxv[i] = px - (float)x0;
            wyv[i] = py - (float)y0;
            const int   x1 = min(x0 + 1, res - 1);
            const int   y1 = min(y0 + 1, res - 1);

            // Wrapping u32 mul/xor, &(2^19-1): bit-exact vs int32 mul^mul % TABLE.
            const uint32_t hx0 = (uint32_t)x0 * HG_HA;
            const uint32_t hx1 = (uint32_t)x1 * HG_HA;
            const uint32_t hy0 = (uint32_t)y0 * HG_HB;
            const uint32_t hy1 = (uint32_t)y1 * HG_HB;

            const float2* __restrict__ t = emb2 + (size_t)l * HG_TABLE;
            e[i][0] = t[(hx0 ^ hy0) & HG_TMASK];
            e[i][1] = t[(hx1 ^ hy0) & HG_TMASK];
            e[i][2] = t[(hx0 ^ hy1) & HG_TMASK];
            e[i][3] = t[(hx1 ^ hy1) & HG_TMASK];
        }

        // --- Phase B: bilinear interp + LDS staging (ds_store_b64) ---
#pragma unroll
        for (int i = 0; i < 4; ++i) {
            const int   l   = g * 4 + i;
            const float wx  = wxv[i], wy = wyv[i];
            const float owx = 1.0f - wx, owy = 1.0f - wy;

            float2 r;
            const float ex0x = e[i][0].x * owx + e[i][1].x * wx;
            const float ex0y = e[i][0].y * owx + e[i][1].y * wx;
            const float ex1x = e[i][2].x * owx + e[i][3].x * wx;
            const float ex1y = e[i][2].y * owx + e[i][3].y * wx;
            r.x = ex0x * owy + ex1x * wy;
            r.y = ex0y * owy + ex1y * wy;

            *(float2*)&lds[tid * HG_STRIDE + 2 * l] = r;   // byte addr 144*tid + 8*l
        }
    }

    __syncthreads();

    // --- CDNA5 async DMA: LDS -> global, 16B/lane, 512B/wave, non-temporal ---
    const uint32_t lds_base = (uint32_t)(uintptr_t)&lds[0];   // as3 offset in low bits
    const size_t   out_base = (size_t)tile * (256u * 32u);    // floats
    const size_t   total    = (size_t)n * 32u;

#pragma unroll
    for (int k = 0; k < 8; ++k) {
        const int j  = k * 256 + tid;    // float4 index within tile (2048 total)
        const int pt = j >> 3;           // point within tile
        const int q  = j & 7;            // float4 within point
        const size_t go = out_base + (size_t)j * 4;
        if (go < total) {
            const uint64_t gaddr = (uint64_t)(uintptr_t)(out + go);
            const uint32_t laddr = lds_base + (uint32_t)(pt * (HG_STRIDE * 4) + q * 16);
            asm volatile("global_store_async_from_lds_b128 %0, %1, off th:TH_STORE_NT"
                         :: "v"(gaddr), "v"(laddr)
                         : "memory");
        }
    }
    // Drain this wave's async LDS->global transfers before LDS teardown.
    asm volatile("s_wait_asynccnt 0x0" ::: "memory");
}

extern "C" void kernel_launch(void* const* d_in, const int* in_sizes, int n_in,
                              void* d_out, int out_size, void* d_ws, size_t ws_size,
                              hipStream_t stream) {
    const float* xy  = (const float*)d_in[0];   // (N, 2) f32
    const float* emb = (const float*)d_in[1];   // (16, 2^19, 2) f32
    float*       out = (float*)d_out;           // (N, 32) f32

    const int n      = in_sizes[0] / 2;         // N points
    const int blocks = (n + 255) / 256;

    hipLaunchKernelGGL(hashgrid2d_encode_kernel, dim3(blocks), dim3(256), 0, stream,
                       xy, emb, out, n);
}